// LSTMModel_2774548873465
// MI455X (gfx1250) — compile-verified
//
#include <hip/hip_runtime.h>
#include <math.h>

// MI455X (gfx1250) LSTM: B=512, T=512, F=128, H=32, C=6
//
// Phase 1 (memory-bound, ~270MB @ 23.3TB/s): xz[t][b][:] = x[b][t][:] @ Wx + b
//   using V_WMMA_F32_16X16X4_F32 (full f32 precision, matches reference).
//   Also emits mask[t][b] = any(x[b][t][:] != 0).
// Phase 2 (latency-bound sequential scan): 32 workgroups x 16 batch rows,
//   8 waves/WG each owning a 16-wide N-tile of the 4H=128 gate columns.
//   Wh fragments register-resident across the t-loop; h/c state in LDS;
//   xz tile loaded directly into the WMMA accumulator. Fused FC+softmax tail.

typedef __attribute__((ext_vector_type(2))) float v2f;
typedef __attribute__((ext_vector_type(8))) float v8f;

#define BSZ 512
#define TSZ 512
#define FSZ 128
#define HSZ 32
#define CSZ 6
#define G4  128   // 4*H

// ---------------------------------------------------------------------------
// Phase 1: xz GEMM + mask.  Grid: (T*B)/16 blocks of 256 threads (8 waves).
// Each block handles a 16-row M-tile (rows share the same t since 512%16==0);
// wave w computes N-tile [w*16, w*16+16).
// ---------------------------------------------------------------------------
__global__ __launch_bounds__(256)
void lstm_xz_gemm(const float* __restrict__ x, const float* __restrict__ Wx,
                  const float* __restrict__ bias, float* __restrict__ xz,
                  float* __restrict__ maskf) {
  __shared__ float xs[16][FSZ + 4];   // staged A tile (16 rows x 128 K)
  __shared__ int rowflag[16];

  const int tid  = threadIdx.x;
  const int lane = tid & 31;
  const int wave = tid >> 5;          // 0..7 -> N-tile
  const int r0   = blockIdx.x * 16;   // global row base, r = t*B + b
  const int t    = r0 >> 9;           // r0 / 512
  const int bb   = r0 & 511;          // batch base

  if (tid < 16) rowflag[tid] = 0;
  __syncthreads();

  // Stage x rows into LDS (each thread: 8 consecutive floats of one row)
  // and reduce the "any nonzero" mask per row.
  {
    const int row = tid >> 4;          // 0..15
    const int cb  = (tid & 15) * 8;    // column base
    const float4* s4 =
        (const float4*)(x + ((size_t)(bb + row) * TSZ + t) * FSZ + cb);
    float4 a = s4[0];
    float4 c = s4[1];
    xs[row][cb + 0] = a.x; xs[row][cb + 1] = a.y;
    xs[row][cb + 2] = a.z; xs[row][cb + 3] = a.w;
    xs[row][cb + 4] = c.x; xs[row][cb + 5] = c.y;
    xs[row][cb + 6] = c.z; xs[row][cb + 7] = c.w;
    bool nz = (a.x != 0.f) | (a.y != 0.f) | (a.z != 0.f) | (a.w != 0.f) |
              (c.x != 0.f) | (c.y != 0.f) | (c.z != 0.f) | (c.w != 0.f);
    if (nz) rowflag[row] = 1;          // benign race: all writers store 1
  }
  __syncthreads();
  if (tid < 16)
    maskf[(size_t)t * BSZ + bb + tid] = rowflag[tid] ? 1.0f : 0.0f;

  // f32 WMMA 16x16x4 fragment indexing (wave32):
  //   A (16x4):  lanes 0-15 hold M=lane, K={k0,k0+1}; lanes 16-31 M=lane-16,
  //              K={k0+2,k0+3}  (2 VGPRs)
  //   B (4x16):  mirrored: lanes 0-15 N=lane K={k0,k0+1}; lanes 16-31 K+=2
  //   C/D (16x16): VGPR r = row r (lanes 0-15) / row r+8 (lanes 16-31),
  //              N = lane%16
  const int n0    = wave * 16;
  const int col   = lane & 15;
  const int koff  = (lane < 16) ? 0 : 2;
  const int rowA  = lane & 15;

  const float bv = bias[n0 + col];
  v8f acc;
#pragma unroll
  for (int r = 0; r < 8; ++r) acc[r] = bv;   // z starts at bias

#pragma unroll 8
  for (int kk = 0; kk < 32; ++kk) {          // K = 128 = 32 steps of 4
    const int k = kk * 4 + koff;
    v2f a;  a.x = xs[rowA][k];               a.y = xs[rowA][k + 1];
    v2f bf; bf.x = Wx[(size_t)k * G4 + n0 + col];
            bf.y = Wx[(size_t)(k + 1) * G4 + n0 + col];
    acc = __builtin_amdgcn_wmma_f32_16x16x4_f32(
        false, a, false, bf, (short)0, acc, false, false);
  }

#pragma unroll
  for (int r = 0; r < 8; ++r) {
    const int row = r + ((lane >= 16) ? 8 : 0);
    xz[((size_t)(r0 + row)) * G4 + n0 + col] = acc[r];
  }
}

// ---------------------------------------------------------------------------
// Phase 2: sequential LSTM scan + FC + softmax.
// Grid: B/16 = 32 blocks of 256 threads (8 waves).
// ---------------------------------------------------------------------------
__global__ __launch_bounds__(256)
void lstm_scan(const float* __restrict__ xz, const float* __restrict__ maskf,
               const float* __restrict__ Wh, const float* __restrict__ Wfc,
               const float* __restrict__ bfc, float* __restrict__ out) {
  __shared__ float hs[16][HSZ + 1];   // padded: conflict-free A-frag loads
  __shared__ float cs[16][HSZ + 1];
  __shared__ float zs[16][G4 + 4];    // z = xz + h@Wh, pitch 132 (bank-safe)
  __shared__ float lg[16][8];         // FC logits

  const int tid    = threadIdx.x;
  const int lane   = tid & 31;
  const int wave   = tid >> 5;
  const int b0     = blockIdx.x * 16;
  const int n0     = wave * 16;
  const int col    = lane & 15;
  const int koff   = (lane < 16) ? 0 : 2;
  const int rowA   = lane & 15;
  const int rshift = (lane >= 16) ? 8 : 0;

  // Wh B-fragments: constant over time -> keep in 16 VGPRs for the whole scan
  v2f bw[8];
#pragma unroll
  for (int kk = 0; kk < 8; ++kk) {           // K = 32 = 8 steps of 4
    const int k = kk * 4 + koff;
    bw[kk].x = Wh[(size_t)k * G4 + n0 + col];
    bw[kk].y = Wh[(size_t)(k + 1) * G4 + n0 + col];
  }

  // h0 = c0 = 0
  for (int i = tid; i < 16 * (HSZ + 1); i += 256) {
    ((float*)hs)[i] = 0.f;
    ((float*)cs)[i] = 0.f;
  }
  __syncthreads();

  for (int t = 0; t < TSZ; ++t) {
    // C accumulator starts at the precomputed input projection xz[t]
    v8f acc;
    const float* xzt = xz + ((size_t)t * BSZ + b0) * G4;
#pragma unroll
    for (int r = 0; r < 8; ++r)
      acc[r] = xzt[(size_t)(r + rshift) * G4 + n0 + col];

    // z += h @ Wh  (8-step f32 WMMA K-chain, A from LDS state)
#pragma unroll
    for (int kk = 0; kk < 8; ++kk) {
      const int k = kk * 4 + koff;
      v2f a; a.x = hs[rowA][k]; a.y = hs[rowA][k + 1];
      acc = __builtin_amdgcn_wmma_f32_16x16x4_f32(
          false, a, false, bw[kk], (short)0, acc, false, false);
    }
#pragma unroll
    for (int r = 0; r < 8; ++r)
      zs[r + rshift][n0 + col] = acc[r];
    __syncthreads();

    // Gate math: 16*32 = 512 elements over 256 threads (uniform, no div.)
#pragma unroll
    for (int e = tid; e < 16 * HSZ; e += 256) {
      const int bi = e >> 5;       // batch row in tile
      const int j  = e & 31;       // hidden unit
      const float zi = zs[bi][j];
      const float zf = zs[bi][j + 32];
      const float zg = zs[bi][j + 64];
      const float zo = zs[bi][j + 96];
      const float ig = 1.f / (1.f + __expf(-zi));
      const float fg = 1.f / (1.f + __expf(-zf));
      const float gg = 1.f - 2.f / (__expf(2.f * zg) + 1.f);  // tanh, inf-safe
      const float og = 1.f / (1.f + __expf(-zo));
      const float cold = cs[bi][j];
      const float cn = fg * cold + ig * gg;
      const float th = 1.f - 2.f / (__expf(2.f * cn) + 1.f);
      const float hn = og * th;
      const float m  = maskf[(size_t)t * BSZ + b0 + bi];
      const float hold = hs[bi][j];
      hs[bi][j] = m * hn + (1.f - m) * hold;   // masked step keeps state
      cs[bi][j] = m * cn + (1.f - m) * cold;
    }
    __syncthreads();
  }

  // FC: logits[b][c] = h @ Wfc + bfc   (16x6, tiny)
  if (tid < 16 * CSZ) {
    const int bi = tid / CSZ, cc = tid % CSZ;
    float a = bfc[cc];
#pragma unroll
    for (int k = 0; k < HSZ; ++k) a += hs[bi][k] * Wfc[k * CSZ + cc];
    lg[bi][cc] = a;
  }
  __syncthreads();

  // softmax over 6 classes, one thread per batch row
  if (tid < 16) {
    float mx = lg[tid][0];
#pragma unroll
    for (int cc = 1; cc < CSZ; ++cc) mx = fmaxf(mx, lg[tid][cc]);
    float ex[CSZ], s = 0.f;
#pragma unroll
    for (int cc = 0; cc < CSZ; ++cc) { ex[cc] = __expf(lg[tid][cc] - mx); s += ex[cc]; }
    const float inv = 1.f / s;
#pragma unroll
    for (int cc = 0; cc < CSZ; ++cc)
      out[(size_t)(b0 + tid) * CSZ + cc] = ex[cc] * inv;
  }
}

// ---------------------------------------------------------------------------
extern "C" void kernel_launch(void* const* d_in, const int* in_sizes, int n_in,
                              void* d_out, int out_size, void* d_ws, size_t ws_size,
                              hipStream_t stream) {
  (void)in_sizes; (void)n_in; (void)out_size; (void)ws_size;
  const float* x   = (const float*)d_in[0];   // [B,T,F]
  const float* Wx  = (const float*)d_in[1];   // [F,4H]
  const float* Wh  = (const float*)d_in[2];   // [H,4H]
  const float* b   = (const float*)d_in[3];   // [4H]
  const float* Wfc = (const float*)d_in[4];   // [H,C]
  const float* bfc = (const float*)d_in[5];   // [C]

  float* xzbuf = (float*)d_ws;                          // [T,B,4H] = 128 MiB
  float* maskf = xzbuf + (size_t)TSZ * BSZ * G4;        // [T,B]    =   1 MiB

  lstm_xz_gemm<<<(TSZ * BSZ) / 16, 256, 0, stream>>>(x, Wx, b, xzbuf, maskf);
  lstm_scan<<<BSZ / 16, 256, 0, stream>>>(xzbuf, maskf, Wh, Wfc, bfc,
                                          (float*)d_out);
}